// Net_40939628265552
// MI455X (gfx1250) — compile-verified
//
#include <hip/hip_runtime.h>

// ---------------------------------------------------------------------------
// SNN forward (snnTorch-style 2-layer LIF) for MI455X / gfx1250.
//
//   cur1 = x @ W1^T + b1                       (once, f32 WMMA 16x16x4)
//   25 steps:
//     mem1 = 0.99*mem1 + cur1 - (mem1>1)       ; spk1 = (mem1>1)
//     cur2 = spk1 @ W2^T + b2                  (f32 WMMA 16x16x4, K=100)
//     mem2 = 0.99*mem2 + cur2 - (mem2>1)       ; spk2 = (mem2>1)
//     record spk2, mem2
//
// One wave32 owns 16 batch rows. All WMMA operands except x are fetched from LDS as
// K-paired float2 (single aligned ds_load_b64 -> lands in an even-aligned
// VGPR pair, no v_mov fixup). Outputs staged through LDS for coalesced
// 128B global stores (the 524 MB of output is the bandwidth bottleneck).
// ---------------------------------------------------------------------------

typedef __attribute__((ext_vector_type(2))) float v2f;
typedef __attribute__((ext_vector_type(8))) float v8f;

#define B_TOTAL   262144
#define IN_F      21
#define HID       100
#define OUT_F     10
#define T_STEPS   25
#define BETA      0.99f
#define THR       1.0f

#define WAVES_PER_BLOCK 4
#define ROWS_PER_WAVE   16
#define ROWS_PER_BLOCK  (WAVES_PER_BLOCK * ROWS_PER_WAVE)   // 64
#define NT1       7      // hidden N-tiles: 7*16 = 112 >= 100
#define K1_STEPS  6      // ceil(21/4): K padded to 24 with zeros
#define K2_STEPS  25     // 100/4 exact
#define SPK_STRIDE 102   // even stride -> 8B-aligned b64 A-fragment reads

__global__ __launch_bounds__(128) void snn_lif_kernel(
    const float* __restrict__ x,    // [B, 21]
    const float* __restrict__ W1,   // [100, 21]
    const float* __restrict__ b1,   // [100]
    const float* __restrict__ W2,   // [10, 100]
    const float* __restrict__ b2,   // [10]
    float* __restrict__ out)        // [25,B,10] spk2 ++ [25,B,10] mem2
{
    // K-paired, transposed, zero-padded weights (block-shared).
    __shared__ v2f  W1p2[12][112];   // W1p2[p][n] = {W1[n][2p], W1[n][2p+1]}
    __shared__ v2f  W2p2[50][16];    // W2p2[p][n] = {W2[n][2p], W2[n][2p+1]}
    __shared__ float b1p[112];
    __shared__ float b2p[16];
    __shared__ float spk[WAVES_PER_BLOCK][ROWS_PER_WAVE][SPK_STRIDE];
    __shared__ float stage[WAVES_PER_BLOCK][2][ROWS_PER_WAVE * OUT_F]; // spk2|mem2

    const int tid  = threadIdx.x;
    const int wave = tid >> 5;
    const int lane = tid & 31;
    const int lo   = lane & 15;   // N index within tile / M index for A-frags
    const int hi   = lane >> 4;   // lane-half selector

    // ---- stage weights into LDS (K-paired, transposed, padded) ----
    for (int i = tid; i < 12 * 112; i += blockDim.x) {
        int p = i / 112, n = i % 112;
        v2f w = {};
        if (n < HID) {
            int k0 = 2 * p;
            w.x = (k0     < IN_F) ? W1[n * IN_F + k0]     : 0.0f;
            w.y = (k0 + 1 < IN_F) ? W1[n * IN_F + k0 + 1] : 0.0f;
        }
        W1p2[p][n] = w;
    }
    for (int i = tid; i < 50 * 16; i += blockDim.x) {
        int p = i / 16, n = i % 16;
        v2f w = {};
        if (n < OUT_F) {
            w.x = W2[n * HID + 2 * p];
            w.y = W2[n * HID + 2 * p + 1];
        }
        W2p2[p][n] = w;
    }
    for (int i = tid; i < 112; i += blockDim.x) b1p[i] = (i < HID)   ? b1[i] : 0.0f;
    for (int i = tid; i < 16;  i += blockDim.x) b2p[i] = (i < OUT_F) ? b2[i] : 0.0f;
    __syncthreads();

    const int rowBase = (blockIdx.x * WAVES_PER_BLOCK + wave) * ROWS_PER_WAVE;

    // ---- GEMM1: cur1[16,100] = x_tile @ W1^T + b1 (f32 WMMA) ----
    v8f cur1[NT1];
    #pragma unroll
    for (int nt = 0; nt < NT1; ++nt) cur1[nt] = {};

    const float* xrow = x + (size_t)(rowBase + lo) * IN_F;  // A: M = lane%16
    #pragma unroll
    for (int kk = 0; kk < K1_STEPS; ++kk) {
        const int k0 = kk * 4 + 2 * hi;                     // K halves 0,1 | 2,3
        v2f a;
        a.x = (k0     < IN_F) ? xrow[k0]     : 0.0f;
        a.y = (k0 + 1 < IN_F) ? xrow[k0 + 1] : 0.0f;
        const int p = 2 * kk + hi;                          // K-pair row
        #pragma unroll
        for (int nt = 0; nt < NT1; ++nt) {
            v2f bm = W1p2[p][nt * 16 + lo];                 // one ds_load_b64
            cur1[nt] = __builtin_amdgcn_wmma_f32_16x16x4_f32(
                false, a, false, bm, (short)0, cur1[nt], false, false);
        }
    }
    #pragma unroll
    for (int nt = 0; nt < NT1; ++nt) {
        const float bb = b1p[nt * 16 + lo];
        #pragma unroll
        for (int r = 0; r < 8; ++r) cur1[nt][r] += bb;
    }

    // ---- LIF state ----
    v8f mem1[NT1];
    #pragma unroll
    for (int nt = 0; nt < NT1; ++nt) mem1[nt] = {};
    v8f mem2 = {};

    float* const outSpk = out;
    float* const outMem = out + (size_t)T_STEPS * B_TOTAL * OUT_F;
    const float bb2 = b2p[lo];

    #pragma unroll 1
    for (int t = 0; t < T_STEPS; ++t) {
        // mem1 update + spike write to LDS (C-frag: row = r+8*hi, col = nt*16+lo)
        #pragma unroll
        for (int nt = 0; nt < NT1; ++nt) {
            const int n = nt * 16 + lo;
            #pragma unroll
            for (int r = 0; r < 8; ++r) {
                float m   = mem1[nt][r];
                float rst = (m > THR) ? THR : 0.0f;
                m = BETA * m + cur1[nt][r] - rst;
                mem1[nt][r] = m;
                if (n < HID)
                    spk[wave][r + 8 * hi][n] = (m > THR) ? 1.0f : 0.0f;
            }
        }
        __syncthreads();   // spikes visible before A-fragment reads

        // ---- GEMM2: cur2[16,10] = spk1 @ W2^T (f32 WMMA, K = 100) ----
        v8f acc = {};
        #pragma unroll
        for (int kk = 0; kk < K2_STEPS; ++kk) {
            const int k0 = kk * 4 + 2 * hi;                 // even -> 8B aligned
            v2f a  = *(const v2f*)&spk[wave][lo][k0];       // one ds_load_b64
            v2f bm = W2p2[2 * kk + hi][lo];                 // one ds_load_b64
            acc = __builtin_amdgcn_wmma_f32_16x16x4_f32(
                false, a, false, bm, (short)0, acc, false, false);
        }

        // mem2 update + stage outputs (only n < 10 is real)
        #pragma unroll
        for (int r = 0; r < 8; ++r) {
            float m   = mem2[r];
            float rst = (m > THR) ? THR : 0.0f;
            m = BETA * m + (acc[r] + bb2) - rst;
            mem2[r] = m;
            if (lo < OUT_F) {
                const int mrow = r + 8 * hi;
                stage[wave][0][mrow * OUT_F + lo] = (m > THR) ? 1.0f : 0.0f;
                stage[wave][1][mrow * OUT_F + lo] = m;
            }
        }
        __syncthreads();   // staging visible before cooperative store

        // coalesced store: 160 floats per tensor per wave (128B bursts)
        const size_t base = ((size_t)t * B_TOTAL + rowBase) * OUT_F;
        #pragma unroll
        for (int i = 0; i < 5; ++i) {
            const int idx = lane + 32 * i;
            outSpk[base + idx] = stage[wave][0][idx];
            outMem[base + idx] = stage[wave][1][idx];
        }
    }
}

extern "C" void kernel_launch(void* const* d_in, const int* in_sizes, int n_in,
                              void* d_out, int out_size, void* d_ws, size_t ws_size,
                              hipStream_t stream) {
    (void)in_sizes; (void)n_in; (void)out_size; (void)d_ws; (void)ws_size;
    const float* x  = (const float*)d_in[0];
    const float* W1 = (const float*)d_in[1];
    const float* b1 = (const float*)d_in[2];
    const float* W2 = (const float*)d_in[3];
    const float* b2 = (const float*)d_in[4];
    float* out = (float*)d_out;

    dim3 grid(B_TOTAL / ROWS_PER_BLOCK);   // 4096 blocks
    dim3 block(WAVES_PER_BLOCK * 32);      // 128 threads = 4 wave32
    snn_lif_kernel<<<grid, block, 0, stream>>>(x, W1, b1, W2, b2, out);
}